// GCN_78142634983506
// MI455X (gfx1250) — compile-verified
//
#include <hip/hip_runtime.h>
#include <math.h>

typedef __attribute__((ext_vector_type(2))) float v2f;
typedef __attribute__((ext_vector_type(8))) float v8f;

#define N_NODES 50000
#define N_EDGES 800000
#define N_QE    100000
#define DIM     128

// ---------------- weight transpose: Wt[n][k] = W[k][n] ----------------
__global__ void k_transpose128(const float* __restrict__ W, float* __restrict__ Wt) {
  int i = blockIdx.x * blockDim.x + threadIdx.x;
  if (i < DIM * DIM) {
    int k = i >> 7, n = i & 127;
    Wt[n * DIM + k] = W[i];
  }
}

// ---------------- accumulator init with broadcast bias (replaces zero+bias pass) ----
__global__ void k_bias_init(float* __restrict__ p, const float* __restrict__ b, int n) {
  int i = blockIdx.x * blockDim.x + threadIdx.x;
  if (i < n) p[i] = b[i & (DIM - 1)];
}

// ---------------- C[M x 128] = act(A)[M x 128] @ W[128 x 128] (+bias, relu) ----------
// Wt is W transposed (Wt[n*128+k] = W[k*128+n]).
// One wave computes a 16x128 strip with V_WMMA_F32_16X16X4_F32, software
// double-buffered: next chunk's A frag + 8 B frags are loaded while the
// current chunk's 8 WMMAs execute (no spills at this tile size).
// A frag (16x4 f32): lanes 0-15 M=0..15 | 16-31 M=0..15; VGPR j holds K=2h+j.
// C/D (16x16 f32): VGPR r -> row r (lanes 0-15) / row r+8 (lanes 16-31), col=lane&15.
template <bool IN_RELU, bool OUT_BIAS_RELU>
__global__ void k_gemm128(const float* __restrict__ A, const float* __restrict__ Wt,
                          float* __restrict__ C, const float* __restrict__ bias,
                          int tiles) {
  int wave = (int)((blockIdx.x * blockDim.x + threadIdx.x) >> 5);
  int lane = threadIdx.x & 31;
  if (wave >= tiles) return;                 // wave-uniform: EXEC stays all-ones
  int m = lane & 15;
  int h = lane >> 4;
  int row0 = wave * 16;

  const float* __restrict__ arow = A + (size_t)(row0 + m) * DIM + 2 * h;
  const float* __restrict__ wcol = Wt + (size_t)m * DIM + 2 * h;

  v8f acc[8];
#pragma unroll
  for (int t = 0; t < 8; ++t) acc[t] = (v8f){0.f,0.f,0.f,0.f,0.f,0.f,0.f,0.f};

  // prologue: chunk 0 fragments
  v2f a_cur = *(const v2f*)(arow);
  if (IN_RELU) { a_cur.x = fmaxf(a_cur.x, 0.f); a_cur.y = fmaxf(a_cur.y, 0.f); }
  v2f b_cur[8];
#pragma unroll
  for (int t = 0; t < 8; ++t) b_cur[t] = *(const v2f*)(wcol + t * 16 * DIM);

#pragma unroll
  for (int k0 = 0; k0 < DIM; k0 += 4) {
    int kn = (k0 + 4) & (DIM - 1);           // wraps to 0 on last iter (discarded)
    v2f a_nxt = *(const v2f*)(arow + kn);
    if (IN_RELU) { a_nxt.x = fmaxf(a_nxt.x, 0.f); a_nxt.y = fmaxf(a_nxt.y, 0.f); }
    v2f b_nxt[8];
#pragma unroll
    for (int t = 0; t < 8; ++t) b_nxt[t] = *(const v2f*)(wcol + t * 16 * DIM + kn);

#pragma unroll
    for (int t = 0; t < 8; ++t)
      acc[t] = __builtin_amdgcn_wmma_f32_16x16x4_f32(
          false, a_cur, false, b_cur[t], (short)0, acc[t], false, false);

    a_cur = a_nxt;
#pragma unroll
    for (int t = 0; t < 8; ++t) b_cur[t] = b_nxt[t];
  }

#pragma unroll
  for (int t = 0; t < 8; ++t) {
    int col = t * 16 + m;
    float bv = OUT_BIAS_RELU ? bias[col] : 0.f;
#pragma unroll
    for (int r = 0; r < 8; ++r) {
      float v = acc[t][r];
      if (OUT_BIAS_RELU) v = fmaxf(v + bv, 0.f);
      C[(size_t)(row0 + r + 8 * h) * DIM + col] = v;
    }
  }
}

// ---------------- out[dst] += H[src], one wave per edge (128 f32 = 512B) ----------------
__global__ void k_aggregate(const float* __restrict__ H, const int* __restrict__ src,
                            const int* __restrict__ dst, float* __restrict__ out, int E) {
  int wave = (int)((blockIdx.x * blockDim.x + threadIdx.x) >> 5);
  int lane = threadIdx.x & 31;
  if (wave >= E) return;
  int s = src[wave];
  int d = dst[wave];
  const float4 v = ((const float4*)(H + (size_t)s * DIM))[lane];
  float* o = out + (size_t)d * DIM + lane * 4;
  atomicAdd(o + 0, v.x);
  atomicAdd(o + 1, v.y);
  atomicAdd(o + 2, v.z);
  atomicAdd(o + 3, v.w);
}

// ---------------- z[e] = emb[src_e] * emb[dst_e], one wave per query edge ----------------
__global__ void k_edge_mul(const float* __restrict__ emb, const int* __restrict__ edges,
                           float* __restrict__ z, int NE) {
  int wave = (int)((blockIdx.x * blockDim.x + threadIdx.x) >> 5);
  int lane = threadIdx.x & 31;
  if (wave >= NE) return;
  int a = edges[2 * wave];
  int b = edges[2 * wave + 1];
  const float4 va = ((const float4*)(emb + (size_t)a * DIM))[lane];
  const float4 vb = ((const float4*)(emb + (size_t)b * DIM))[lane];
  float4 r;
  r.x = va.x * vb.x; r.y = va.y * vb.y; r.z = va.z * vb.z; r.w = va.w * vb.w;
  ((float4*)(z + (size_t)wave * DIM))[lane] = r;
}

// ---------------- tail: z@P3 + pb3, L2-normalize, log_softmax ----------------
__global__ void k_tail(const float* __restrict__ z, const float* __restrict__ P3,
                       const float* __restrict__ pb3, float* __restrict__ out, int NE) {
  int r = blockIdx.x * blockDim.x + threadIdx.x;
  if (r >= NE) return;
  const float* zr = z + (size_t)r * DIM;
  float d0 = pb3[0], d1 = pb3[1];
#pragma unroll 8
  for (int i = 0; i < DIM; ++i) {
    float v = zr[i];
    d0 += v * P3[2 * i];
    d1 += v * P3[2 * i + 1];
  }
  float nrm = fmaxf(sqrtf(d0 * d0 + d1 * d1), 1e-12f);
  float a = d0 / nrm, b = d1 / nrm;
  float mx = fmaxf(a, b);
  float lse = mx + logf(expf(a - mx) + expf(b - mx));
  out[2 * r + 0] = a - lse;
  out[2 * r + 1] = b - lse;
}

extern "C" void kernel_launch(void* const* d_in, const int* in_sizes, int n_in,
                              void* d_out, int out_size, void* d_ws, size_t ws_size,
                              hipStream_t stream) {
  const float* x   = (const float*)d_in[0];
  const int*   adj = (const int*)d_in[1];        // [2,E] row-major
  const int*   te  = (const int*)d_in[2];        // [NE,2] row-major
  const float* W1  = (const float*)d_in[3];  const float* b1  = (const float*)d_in[4];
  const float* W2  = (const float*)d_in[5];  const float* b2  = (const float*)d_in[6];
  const float* W3  = (const float*)d_in[7];  const float* b3  = (const float*)d_in[8];
  const float* P1  = (const float*)d_in[9];  const float* pb1 = (const float*)d_in[10];
  const float* P2  = (const float*)d_in[11]; const float* pb2 = (const float*)d_in[12];
  const float* P3  = (const float*)d_in[13]; const float* pb3 = (const float*)d_in[14];
  float* out = (float*)d_out;

  const int* src = adj;            // edge_index[0]
  const int* dst = adj + N_EDGES;  // edge_index[1]

  // workspace layout: two NE*128 ping-pong regions + 5 transposed 128x128 weights
  float* U0  = (float*)d_ws;                          // NE*DIM floats
  float* U1  = U0 + (size_t)N_QE * DIM;               // NE*DIM floats
  float* Wt1 = U1 + (size_t)N_QE * DIM;
  float* Wt2 = Wt1 + DIM * DIM;
  float* Wt3 = Wt2 + DIM * DIM;
  float* WtP1 = Wt3 + DIM * DIM;
  float* WtP2 = WtP1 + DIM * DIM;

  const int TB = 256;
  const int tgrid = (DIM * DIM + TB - 1) / TB;
  k_transpose128<<<tgrid, TB, 0, stream>>>(W1, Wt1);
  k_transpose128<<<tgrid, TB, 0, stream>>>(W2, Wt2);
  k_transpose128<<<tgrid, TB, 0, stream>>>(W3, Wt3);
  k_transpose128<<<tgrid, TB, 0, stream>>>(P1, WtP1);
  k_transpose128<<<tgrid, TB, 0, stream>>>(P2, WtP2);

  const int tilesN  = N_NODES / 16;  // 3125
  const int tilesNE = N_QE / 16;     // 6250
  const int gemmGridN  = (tilesN  + 7) / 8;   // 8 waves per 256-thread block
  const int gemmGridNE = (tilesNE + 7) / 8;
  const int nElemN  = N_NODES * DIM;
  const int initGrid = (nElemN + TB - 1) / TB;
  const int aggGrid = N_EDGES / 8;          // one wave per edge, 8 waves per block
  const int emGrid  = N_QE / 8;

  // ---- GCN layer 1: h1 = relu-deferred( agg(x@W1) + b1 ) ----
  k_gemm128<false, false><<<gemmGridN, TB, 0, stream>>>(x, Wt1, U0, nullptr, tilesN);
  k_bias_init<<<initGrid, TB, 0, stream>>>(U1, b1, nElemN);
  k_aggregate<<<aggGrid, TB, 0, stream>>>(U0, src, dst, U1, N_EDGES);

  // ---- GCN layer 2 (relu of layer-1 folded into A-fragment load) ----
  k_gemm128<true, false><<<gemmGridN, TB, 0, stream>>>(U1, Wt2, U0, nullptr, tilesN);
  k_bias_init<<<initGrid, TB, 0, stream>>>(U1, b2, nElemN);
  k_aggregate<<<aggGrid, TB, 0, stream>>>(U0, src, dst, U1, N_EDGES);

  // ---- GCN layer 3 (relu of layer-2 folded in; emb = U1, no output relu) ----
  k_gemm128<true, false><<<gemmGridN, TB, 0, stream>>>(U1, Wt3, U0, nullptr, tilesN);
  k_bias_init<<<initGrid, TB, 0, stream>>>(U1, b3, nElemN);
  k_aggregate<<<aggGrid, TB, 0, stream>>>(U0, src, dst, U1, N_EDGES);

  // ---- LinkPredictor (bias+relu folded into GEMM epilogues) ----
  k_edge_mul<<<emGrid, TB, 0, stream>>>(U1, te, U0, N_QE);
  k_gemm128<false, true><<<gemmGridNE, TB, 0, stream>>>(U0, WtP1, U1, pb1, tilesNE);
  k_gemm128<false, true><<<gemmGridNE, TB, 0, stream>>>(U1, WtP2, U0, pb2, tilesNE);
  k_tail<<<(N_QE + TB - 1) / TB, TB, 0, stream>>>(U0, P3, pb3, out, N_QE);
}